// EBSDCovmatKSphere_39968965656982
// MI455X (gfx1250) — compile-verified
//
#include <hip/hip_runtime.h>

typedef __attribute__((ext_vector_type(2))) float v2f;
typedef __attribute__((ext_vector_type(8))) float v8f;

// ---------------------------------------------------------------------------
// Stage 1a: A-matrix rows. For row r = n*2G + k:
//   e = rot(q_{k%G}, query_n) * (k<G ? +1 : -1);  E[r] = (ex, ey, ez, |e|^2)
// ---------------------------------------------------------------------------
__global__ void build_rows_kernel(const float* __restrict__ quats,
                                  const float* __restrict__ query,
                                  float4* __restrict__ E,
                                  int N, int G) {
  int idx = blockIdx.x * blockDim.x + threadIdx.x;
  int K2 = 2 * G;
  if (idx >= N * K2) return;
  int n = idx / K2;
  int k = idx - n * K2;
  int g = (k < G) ? k : (k - G);
  float qw = quats[g * 4 + 0], qx = quats[g * 4 + 1];
  float qy = quats[g * 4 + 2], qz = quats[g * 4 + 3];
  float vx = query[n * 3 + 0], vy = query[n * 3 + 1], vz = query[n * 3 + 2];
  // t = 2 * cross(qv, v);  e = v + qw*t + cross(qv, t)
  float tx = 2.0f * (qy * vz - qz * vy);
  float ty = 2.0f * (qz * vx - qx * vz);
  float tz = 2.0f * (qx * vy - qy * vx);
  float ex = vx + qw * tx + (qy * tz - qz * ty);
  float ey = vy + qw * ty + (qz * tx - qx * tz);
  float ez = vz + qw * tz + (qx * ty - qy * tx);
  if (k >= G) { ex = -ex; ey = -ey; ez = -ez; }
  float e2 = ex * ex + ey * ey + ez * ez;
  E[idx] = make_float4(ex, ey, ez, e2);
}

// ---------------------------------------------------------------------------
// Stage 1b: B-matrix columns + per-column accumulator init:
//   Bp[m] = (-2 sx, -2 sy, -2 sz, 1.0);  s2n[m] = |s|^2
// ---------------------------------------------------------------------------
__global__ void build_cols_kernel(const float* __restrict__ s2pts,
                                  float4* __restrict__ Bp,
                                  float* __restrict__ s2n, int M) {
  int m = blockIdx.x * blockDim.x + threadIdx.x;
  if (m >= M) return;
  float sx = s2pts[m * 3 + 0], sy = s2pts[m * 3 + 1], sz = s2pts[m * 3 + 2];
  Bp[m] = make_float4(-2.0f * sx, -2.0f * sy, -2.0f * sz, 1.0f);
  s2n[m] = sx * sx + sy * sy + sz * sz;
}

// ---------------------------------------------------------------------------
// Monotonic f32 -> u32 key: preserves total order of floats (incl. negatives).
// ---------------------------------------------------------------------------
__device__ __forceinline__ unsigned int f32_order_key(float f) {
  unsigned int u = __float_as_uint(f);
  return u ^ ((unsigned int)((int)u >> 31) | 0x80000000u);
}

// ---------------------------------------------------------------------------
// Stage 2: 4 waves (128 threads) per query n; wave w owns col tiles
// ct = w, w+4, ... (wrap-around; min-reduce is idempotent).  d2 tile via
// V_WMMA_F32_16X16X4_F32; branchless argmin by 64-bit packed key
// ((orderkey(d2)<<32)|flat) so u64-min == lexicographic (d2, flat) min with
// first-occurrence tie-break.  Scalar-uniform trip count + readfirstlane'd
// wave id keep loop control in SALU (no EXEC writes around the WMMA).
// LDS reduce, ids[n] = flat % M.
// ---------------------------------------------------------------------------
__global__ void __launch_bounds__(128)
argmin_wmma_kernel(const float4* __restrict__ E,
                   const float4* __restrict__ Bp,
                   const float* __restrict__ s2n,
                   int* __restrict__ ids,
                   int M, int K2) {
  const int n = blockIdx.x;
  const int tid = threadIdx.x;
  const int lane = tid & 31;
  const int half = lane >> 4;     // 0: K={0,1} / rows 0..7 ; 1: K={2,3} / rows 8..15
  const int l16 = lane & 15;      // A-row / B-col / D-col within tile
  // Wave id, broadcast to a scalar so loop/address math stays in SALU.
  const int wv = __builtin_amdgcn_readfirstlane(tid >> 5);   // 0..3

  const int rowTiles = (K2 + 15) >> 4;   // 3 for K2=48
  const int colTiles = M >> 4;           // 256 for M=4096 (assumed >= 4)
  const int iters = (colTiles + 3) >> 2; // uniform scalar trip count

  unsigned long long best = ~0ull;

  for (int kt = 0; kt < rowTiles; ++kt) {
    int row = n * K2 + kt * 16 + l16;
    float4 a4 = E[row];                       // exact tiling: always in range
    v2f A;
    A[0] = half ? a4.z : a4.x;                // K = 2*half + 0
    A[1] = half ? a4.w : a4.y;                // K = 2*half + 1
    int kbase = (kt * 16 + half * 8) * M;     // flat base for j = 0

    for (int it = 0; it < iters; ++it) {
      int ct = wv + it * 4;                   // scalar
      ct = (ct < colTiles) ? ct : (ct - colTiles);  // wrap: duplicate is harmless
      int col = ct * 16 + l16;
      float4 b4 = Bp[col];
      v2f B;
      B[0] = half ? b4.z : b4.x;
      B[1] = half ? b4.w : b4.y;
      float s2 = s2n[col];
      v8f C;
#pragma unroll
      for (int j = 0; j < 8; ++j) C[j] = s2;  // C[row][col] = |s_col|^2

      // d2 tile: D = A x B + C  == |e|^2 + |s|^2 - 2 e.s  (exact reference d2)
      v8f D = __builtin_amdgcn_wmma_f32_16x16x4_f32(
          false, A, false, B, (short)0, C, false, false);

#pragma unroll
      for (int j = 0; j < 8; ++j) {
        unsigned int fl = (unsigned int)(kbase + j * M + col);   // k*M + m
        unsigned long long key =
            ((unsigned long long)f32_order_key(D[j]) << 32) | fl;
        best = (key < best) ? key : best;     // branchless v_min_u64
      }
    }
  }

  __shared__ unsigned long long sK[128];
  sK[tid] = best;
  __syncthreads();
  if (tid == 0) {
    unsigned long long b = sK[0];
    for (int i = 1; i < 128; ++i) b = (sK[i] < b) ? sK[i] : b;
    ids[n] = (int)((unsigned int)b % (unsigned int)M);  // low 32 bits = flat
  }
}

// ---------------------------------------------------------------------------
// Stage 3: out[i][j] = covmat[ids[i]][ids[j]].  ids[i] is block-uniform ->
// scalar load; consecutive j gathers hit one covmat row (L2 resident).
// ---------------------------------------------------------------------------
__global__ void gather_kernel(const float* __restrict__ covmat,
                              const int* __restrict__ ids,
                              float* __restrict__ out, int N, int M) {
  int i = blockIdx.x;                           // uniform per block
  int j = blockIdx.y * blockDim.x + threadIdx.x;
  if (j >= N) return;
  int ri = ids[i];
  int cj = ids[j];
  out[i * N + j] = covmat[(long)ri * M + cj];
}

// ---------------------------------------------------------------------------
extern "C" void kernel_launch(void* const* d_in, const int* in_sizes, int n_in,
                              void* d_out, int out_size, void* d_ws, size_t ws_size,
                              hipStream_t stream) {
  const float* query = (const float*)d_in[0];   // (N,3)
  const float* s2pts = (const float*)d_in[1];   // (M,3)
  const float* quats = (const float*)d_in[2];   // (G,4)
  const float* covmat = (const float*)d_in[3];  // (M,M)
  float* out = (float*)d_out;

  const int N = in_sizes[0] / 3;   // 1024
  const int M = in_sizes[1] / 3;   // 4096
  const int G = in_sizes[2] / 4;   // 24
  const int K2 = 2 * G;            // 48
  const int rows = N * K2;         // 49152

  // Workspace layout (16B aligned): E | Bp | s2n | ids
  char* ws = (char*)d_ws;
  float4* E = (float4*)ws;                        size_t off = (size_t)rows * 16;
  float4* Bp = (float4*)(ws + off);               off += (size_t)M * 16;
  float* s2n = (float*)(ws + off);                off += (size_t)M * 4;
  int* ids = (int*)(ws + off);

  build_rows_kernel<<<(rows + 255) / 256, 256, 0, stream>>>(quats, query, E, N, G);
  build_cols_kernel<<<(M + 255) / 256, 256, 0, stream>>>(s2pts, Bp, s2n, M);
  argmin_wmma_kernel<<<N, 128, 0, stream>>>(E, Bp, s2n, ids, M, K2);

  dim3 ggrid(N, (N + 255) / 256);
  gather_kernel<<<ggrid, 256, 0, stream>>>(covmat, ids, out, N, M);
}